// grad_gelu_15676630630555
// MI455X (gfx1250) — compile-verified
//
#include <hip/hip_runtime.h>
#include <hip/hip_bf16.h>

// grad of GELU (exact-erf form):
//   out = Phi(x) + x * phi(x)
//   Phi(x) = 0.5*(1+erf(x/sqrt(2))),  phi(x) = inv_sqrt(2*pi)*exp(-x^2/2)
//
// Key trick: with z = |x|/sqrt(2), exp(-z*z) == exp(-0.5*x*x), so a single
// v_exp_f32 feeds both the pdf and the Abramowitz-Stegun erf polynomial
// (max abs err ~1.5e-7 in erf -> ~7.5e-8 in Phi).

// Native clang vector type: accepted by __builtin_nontemporal_{load,store}
// and lowers to global_load_b128 / global_store_b128.
typedef float vf4 __attribute__((ext_vector_type(4)));

__device__ __forceinline__ float gelu_grad(float x) {
    const float kInvSqrt2   = 0.70710678118654752f;
    const float kInvSqrt2Pi = 0.39894228040143267f;

    float z = __builtin_fabsf(x) * kInvSqrt2;
    float e = __expf(-z * z);                      // v_exp_f32 (one trans op)

    // Abramowitz & Stegun 7.1.26: erf(z) = 1 - poly(t)*exp(-z^2), t = 1/(1+p*z)
    float t = __builtin_amdgcn_rcpf(__builtin_fmaf(0.3275911f, z, 1.0f)); // v_rcp_f32
    float p = __builtin_fmaf(1.061405429f, t, -1.453152027f);
    p = __builtin_fmaf(p, t, 1.421413741f);
    p = __builtin_fmaf(p, t, -0.284496736f);
    p = __builtin_fmaf(p, t, 0.254829592f);
    float erf_abs = __builtin_fmaf(-p * t, e, 1.0f);

    // erf(-z) = -erf(z): transfer sign of x
    float erfv = __builtin_copysignf(erf_abs, x);

    float cdf = __builtin_fmaf(0.5f, erfv, 0.5f);  // 0.5*(1+erf)
    return __builtin_fmaf(x, kInvSqrt2Pi * e, cdf); // cdf + x*pdf
}

// Main vectorized kernel: one 16B vf4 per thread per loop trip.
// Non-temporal load+store: 537 MB streamed > 192 MB L2, zero reuse.
__global__ __launch_bounds__(256) void grad_gelu_v4_kernel(
    const vf4* __restrict__ in, vf4* __restrict__ out, int n4) {
    int idx    = blockIdx.x * blockDim.x + threadIdx.x;
    int stride = gridDim.x * blockDim.x;
    for (int i = idx; i < n4; i += stride) {
        vf4 v = __builtin_nontemporal_load(&in[i]);   // global_load_b128 TH=NT
        vf4 r;
        r.x = gelu_grad(v.x);
        r.y = gelu_grad(v.y);
        r.z = gelu_grad(v.z);
        r.w = gelu_grad(v.w);
        __builtin_nontemporal_store(r, &out[i]);      // global_store_b128 TH=NT
    }
}

// Scalar tail for n % 4 (not hit for 16*2048*2048, but keep it deterministic
// and correct for any size).
__global__ __launch_bounds__(64) void grad_gelu_tail_kernel(
    const float* __restrict__ in, float* __restrict__ out, int base, int n) {
    int i = base + blockIdx.x * blockDim.x + threadIdx.x;
    if (i < n) {
        out[i] = gelu_grad(__builtin_nontemporal_load(&in[i]));
    }
}

extern "C" void kernel_launch(void* const* d_in, const int* in_sizes, int n_in,
                              void* d_out, int out_size, void* d_ws, size_t ws_size,
                              hipStream_t stream) {
    (void)n_in; (void)d_ws; (void)ws_size; (void)out_size;
    const float* x = (const float*)d_in[0];
    float* out = (float*)d_out;
    int n  = in_sizes[0];          // 16*2048*2048 = 67,108,864
    int n4 = n >> 2;               // vf4 count
    int rem = n & 3;

    if (n4 > 0) {
        const int block = 256;     // 8 wave32 per workgroup
        long long want = ((long long)n4 + block - 1) / block;
        int grid = (int)(want > 2147483647LL ? 2147483647LL : want);
        grad_gelu_v4_kernel<<<grid, block, 0, stream>>>(
            (const vf4*)x, (vf4*)out, n4);
    }
    if (rem > 0) {
        grad_gelu_tail_kernel<<<1, 64, 0, stream>>>(x, out, n4 << 2, n);
    }
}